// AtomFeature_90683939487976
// MI455X (gfx1250) — compile-verified
//
#include <hip/hip_runtime.h>
#include <hip/hip_bf16.h>

// CDNA5 / gfx1250 — wave32. WMMA f32 16x16x4 used for the Gram-matrix tile
// of the kNN distance computation. 16x4800 distance strip held in LDS
// (307 KB; fits CDNA5's 320 KB-per-workgroup LDS).

typedef float v2f __attribute__((ext_vector_type(2)));
typedef float v8f __attribute__((ext_vector_type(8)));

#define EMB_DIM 128
#define NTYPES  12
#define KSEL    30
#define SEPS    1e-6f
#define LEPS    1e6f

// ---------------------------------------------------------------------------
// Kernel 1: atom embedding (cyclic types) + graph norm.
// One block per (channel d, batch b); block-reduces sum / sumsq / count over N,
// then streams the normalized output.
// var = (1/cnt) * sum_n (mf - mean)^2  over ALL n  (masked-out lanes contribute
// mean^2), expanded as (S2 - 2*mean*S1 + N*mean^2)/cnt.
// ---------------------------------------------------------------------------
__global__ __launch_bounds__(256) void emb_graphnorm_kernel(
    const float* __restrict__ emb_w,   // (12,128)
    const float* __restrict__ mask,    // (B,N)
    const float* __restrict__ scale,   // (128)
    const float* __restrict__ shift,   // (128)
    float* __restrict__ out,           // (B,N,128)
    int B, int N)
{
    const int d   = blockIdx.x;        // channel
    const int b   = blockIdx.y;        // batch
    const int tid = threadIdx.x;

    __shared__ float s1a[256], s2a[256], sca[256];

    float s1 = 0.f, s2 = 0.f, c = 0.f;
    for (int n = tid; n < N; n += 256) {
        float w    = emb_w[(n % NTYPES) * EMB_DIM + d];
        float m    = mask[(size_t)b * N + n];
        float feat = w * m;        // emb after initial mask multiply
        float mf   = feat * m;     // graph-norm masks again
        s1 += mf;
        s2 += mf * mf;
        c  += m;
    }
    s1a[tid] = s1; s2a[tid] = s2; sca[tid] = c;
    __syncthreads();
    for (int off = 128; off > 0; off >>= 1) {
        if (tid < off) {
            s1a[tid] += s1a[tid + off];
            s2a[tid] += s2a[tid + off];
            sca[tid] += sca[tid + off];
        }
        __syncthreads();
    }
    float cnt = sca[0];
    if (cnt == 0.f) cnt = 1.f;
    float S1   = s1a[0], S2 = s2a[0];
    float mean = S1 / cnt;
    float var  = (S2 - 2.f * mean * S1 + (float)N * mean * mean) / cnt;
    float istd = rsqrtf(var + SEPS);
    float sc   = scale[d], sh = shift[d];

    for (int n = tid; n < N; n += 256) {
        float w    = emb_w[(n % NTYPES) * EMB_DIM + d];
        float m    = mask[(size_t)b * N + n];
        float feat = w * m;
        out[((size_t)b * N + n) * EMB_DIM + d] =
            ((feat - mean) * istd * sc + sh) * m;
    }
}

// ---------------------------------------------------------------------------
// Kernel 2: fused pairwise-distance + top-K(=30) smallest per row.
// One workgroup (8 waves, 256 threads) per (batch, 16-row query tile).
//   Stage 1: each wave computes 16x16 distance tiles with
//            V_WMMA_F32_16X16X4_F32 (A = 16x4 query coords, B = 4x16 cand
//            coords, K=3 padded to 4) and writes the 16xN strip to LDS.
//            Epilogue uses raw v_sqrt_f32 (__builtin_amdgcn_sqrtf) to avoid
//            the IEEE software-refinement expansion (~1 ulp is ample for
//            distances feeding a top-k).
//   Stage 2: 2 rows per wave; 30 rounds of wave32 argmin over the LDS row
//            (shfl_xor reduce, tie -> lower index to match lax.top_k),
//            selected entry replaced with +BIG.
// The (B,N,N) distance matrix never touches HBM.
// ---------------------------------------------------------------------------
__global__ __launch_bounds__(256) void knn_wmma_kernel(
    const float* __restrict__ coords,  // (B,N,3)
    const float* __restrict__ mask,    // (B,N)
    float* __restrict__ out_d,         // (B,N,KSEL)
    int*   __restrict__ out_i,         // (B,N,KSEL)
    int B, int N)
{
    extern __shared__ float ldsDist[]; // [16 * N]
    __shared__ float s_nq[16];         // query squared norms
    __shared__ float s_qm[16];         // query masks

    const int b    = blockIdx.y;
    const int i0   = blockIdx.x * 16;
    const int tid  = threadIdx.x;
    const int lane = tid & 31;
    const int wave = tid >> 5;

    const float* C = coords + (size_t)b * N * 3;
    const float* M = mask   + (size_t)b * N;

    // ---- load the A tile (16 query rows) into per-wave WMMA registers ----
    // f32 A 16x4 layout: lanes 0-15 hold (K=0,K=1) of row M=lane,
    //                    lanes 16-31 hold (K=2,K=3) of row M=lane-16.
    const int qrow = lane & 15;
    const int qi   = i0 + qrow;
    float qx = 0.f, qy = 0.f, qz = 0.f, qm = 0.f;
    if (qi < N) {
        qx = C[qi * 3 + 0];
        qy = C[qi * 3 + 1];
        qz = C[qi * 3 + 2];
        qm = M[qi];
    }
    v2f amat;
    if (lane < 16) { amat.x = qx; amat.y = qy; }
    else           { amat.x = qz; amat.y = 0.f; }
    if (wave == 0 && lane < 16) {
        s_nq[lane] = qx * qx + qy * qy + qz * qz;
        s_qm[lane] = qm;
    }
    __syncthreads();

    // ---- stage 1: distance strip via WMMA ----
    const int nTiles = (N + 15) >> 4;
    for (int jt = wave; jt < nTiles; jt += 8) {
        const int j = jt * 16 + (lane & 15);
        float cx = 0.f, cy = 0.f, cz = 0.f, cm = 0.f;
        if (j < N) {
            cx = C[j * 3 + 0];
            cy = C[j * 3 + 1];
            cz = C[j * 3 + 2];
            cm = M[j];
        }
        const float nc = cx * cx + cy * cy + cz * cz;

        // f32 B 4x16 layout mirrors A: lanes 0-15 carry (K=0,K=1) of col N=lane,
        // lanes 16-31 carry (K=2,K=3) of col N=lane-16.
        v2f bmat;
        if (lane < 16) { bmat.x = cx; bmat.y = cy; }
        else           { bmat.x = cz; bmat.y = 0.f; }

        v8f acc = {};
        // D = A x B (+0): 16x16 tile of q.c dot products
        acc = __builtin_amdgcn_wmma_f32_16x16x4_f32(
            /*neg_a=*/false, amat, /*neg_b=*/false, bmat,
            /*c_mod=*/(short)0, acc, /*reuse_a=*/false, /*reuse_b=*/false);

        const int hi = (lane >= 16) ? 8 : 0;
        #pragma unroll
        for (int vv = 0; vv < 8; ++vv) {
            const int m = vv + hi;                // row within the 16-row tile
            float dot = acc[vv];
            float d2  = s_nq[m] + nc - 2.f * dot + SEPS;
            // raw v_sqrt_f32 (no software refinement sequence)
            float dd  = __builtin_amdgcn_sqrtf(fmaxf(d2, 0.f));
            if (i0 + m == j) dd += LEPS;          // eye * LEPS
            float m2 = s_qm[m] * cm;              // mask2d
            dd = dd * m2 + (1.f - m2) * LEPS;
            if (j < N) ldsDist[(size_t)m * N + j] = dd;
        }
    }
    __syncthreads();

    // ---- stage 2: top-K smallest per row (2 rows per wave) ----
    const float BIG = 3.0e38f;
    for (int rr = 0; rr < 2; ++rr) {
        const int r  = wave * 2 + rr;
        const int gi = i0 + r;
        float* row = ldsDist + (size_t)r * N;
        for (int k = 0; k < KSEL; ++k) {
            float v  = BIG;
            int   ix = 0x7fffffff;
            for (int t = lane; t < N; t += 32) {
                float x = row[t];
                if (x < v || (x == v && t < ix)) { v = x; ix = t; }
            }
            // wave32 min-reduce, tie -> lower index (matches lax.top_k order)
            #pragma unroll
            for (int off = 16; off > 0; off >>= 1) {
                float ov = __shfl_xor(v, off, 32);
                int   oi = __shfl_xor(ix, off, 32);
                if (ov < v || (ov == v && oi < ix)) { v = ov; ix = oi; }
            }
            if (lane == 0 && gi < N) {
                float od  = v;
                int   oix = (ix == gi) ? -1 : ix;   // self-edge -> -1
                if (s_qm[r] == 0.f) { od = LEPS; oix = -1; }  // padded query
                out_d[((size_t)b * N + gi) * KSEL + k] = od;
                out_i[((size_t)b * N + gi) * KSEL + k] = oix;
            }
            if (lane == (ix & 31)) row[ix] = BIG;   // exclude for next round
            __syncthreads();                        // uniform across all waves
        }
    }
}

// ---------------------------------------------------------------------------
extern "C" void kernel_launch(void* const* d_in, const int* in_sizes, int n_in,
                              void* d_out, int out_size, void* d_ws, size_t ws_size,
                              hipStream_t stream) {
    const float* coords = (const float*)d_in[0];   // (B,N,3)
    const float* amask  = (const float*)d_in[1];   // (B,N)
    const float* embw   = (const float*)d_in[2];   // (12,128)
    const float* scale  = (const float*)d_in[3];   // (128)
    const float* shift  = (const float*)d_in[4];   // (128)

    const int B = 2;
    const int N = in_sizes[1] / B;                 // 4800

    float* out      = (float*)d_out;
    float* out_emb  = out;                                          // B*N*128
    float* out_dist = out + (size_t)B * N * EMB_DIM;                // B*N*30
    int*   out_idx  = (int*)(out + (size_t)B * N * EMB_DIM + (size_t)B * N * KSEL);

    // Kernel 1: embedding + graph norm (one block per (channel, batch))
    dim3 g1(EMB_DIM, B);
    emb_graphnorm_kernel<<<g1, 256, 0, stream>>>(embw, amask, scale, shift,
                                                 out_emb, B, N);

    // Kernel 2: fused WMMA distances + top-K.  16*N floats of dynamic LDS
    // (307 KB for N=4800 — within CDNA5's 320 KB per-workgroup LDS).
    dim3 g2((N + 15) / 16, B);
    size_t shmem = (size_t)16 * N * sizeof(float);
    knn_wmma_kernel<<<g2, 256, shmem, stream>>>(coords, amask,
                                                out_dist, out_idx, B, N);
}